// CellularGNN_5488968204640
// MI455X (gfx1250) — compile-verified
//
#include <hip/hip_runtime.h>
#include <hip/hip_bf16.h>

// ---------------------------------------------------------------------------
// GCN forward on MI455X (gfx1250, wave32).
//  - Dense GEMMs use V_WMMA_F32_16X16X4_F32 (f32 matrix path, exact precision).
//  - Edge scatter uses wave-per-edge with hardware f32 global atomics; h1/h2
//    working sets (12.8/25.6 MB) are L2-resident on the 192MB L2.
//  - GEMM epilogue: wave-uniform full-tile fast path (no per-row exec
//    masking) since M = 100000 = 6250 exact 16-row tiles.
// ---------------------------------------------------------------------------

typedef float v2f __attribute__((ext_vector_type(2)));
typedef float v8f __attribute__((ext_vector_type(8)));

#define IN_CH 128

// ---------------- degree / normalization ----------------

__global__ void k_init_deg(float* deg, int n) {
    int i = blockIdx.x * blockDim.x + threadIdx.x;
    if (i < n) deg[i] = 1.0f;  // self-loop contributes 1
}

__global__ void k_deg_accum(const int* __restrict__ dst, float* deg, int E) {
    int e = blockIdx.x * blockDim.x + threadIdx.x;
    if (e < E) unsafeAtomicAdd(&deg[dst[e]], 1.0f);
}

__global__ void k_dinv(const float* __restrict__ deg, float* dinv, int n) {
    int i = blockIdx.x * blockDim.x + threadIdx.x;
    if (i < n) {
        float d = deg[i];
        dinv[i] = d > 0.0f ? __frsqrt_rn(d) : 0.0f;
    }
}

// ---------------- WMMA GEMM: O[M,N] = A[M,K] @ W[K,N] ----------------
// One wave computes one 16xN row-tile. K multiple of 4, N multiple of 16.

template <int K, int N>
__global__ void k_gemm_wmma(const float* __restrict__ A,
                            const float* __restrict__ W,
                            float* __restrict__ O,
                            int n_rows, int n_tiles) {
    int wave = (blockIdx.x * blockDim.x + threadIdx.x) >> 5;
    int lane = threadIdx.x & 31;
    if (wave >= n_tiles) return;               // whole-wave uniform guard

    const int m0    = wave * 16;
    const int mrow  = m0 + (lane & 15);
    const int rm    = mrow < n_rows ? mrow : (n_rows - 1);   // clamp reads
    const int khalf = (lane >> 4) * 2;                       // 0 or 2
    const int colL  = lane & 15;

    constexpr int NT = N / 16;
    v8f acc[NT] = {};

    for (int k = 0; k < K; k += 4) {
        v2f a;
        a.x = A[rm * K + k + khalf];
        a.y = A[rm * K + k + khalf + 1];
#pragma unroll
        for (int t = 0; t < NT; ++t) {
            v2f b;
            b.x = W[(k + khalf) * N + t * 16 + colL];
            b.y = W[(k + khalf + 1) * N + t * 16 + colL];
            acc[t] = __builtin_amdgcn_wmma_f32_16x16x4_f32(
                false, a, false, b, (short)0, acc[t], false, false);
        }
    }

    // C/D layout: VGPR r -> M = r (lanes 0-15) / r+8 (lanes 16-31), N = lane&15
    const int mbase = m0 + (lane >> 4) * 8;
    if (m0 + 16 <= n_rows) {
        // fast path: full tile, no predication (wave-uniform branch)
#pragma unroll
        for (int t = 0; t < NT; ++t) {
#pragma unroll
            for (int r = 0; r < 8; ++r) {
                O[(mbase + r) * N + t * 16 + colL] = acc[t][r];
            }
        }
    } else {
#pragma unroll
        for (int t = 0; t < NT; ++t) {
#pragma unroll
            for (int r = 0; r < 8; ++r) {
                int row = mbase + r;
                if (row < n_rows) O[row * N + t * 16 + colL] = acc[t][r];
            }
        }
    }
}

// ---------------- self-loop init: out = dinv[n]^2 * h ----------------

template <int LOGC>
__global__ void k_selfloop(const float* __restrict__ h,
                           const float* __restrict__ dinv,
                           float* __restrict__ out, int total) {
    int id = blockIdx.x * blockDim.x + threadIdx.x;
    if (id >= total) return;
    float d = dinv[id >> LOGC];
    out[id] = d * d * h[id];
}

// ---------------- edge scatter (wave per edge) ----------------

__global__ void k_edge_scatter32(const int* __restrict__ src,
                                 const int* __restrict__ dst,
                                 const float* __restrict__ dinv,
                                 const float* __restrict__ h,
                                 float* out, int E) {
    int w = (blockIdx.x * blockDim.x + threadIdx.x) >> 5;
    int lane = threadIdx.x & 31;
    if (w >= E) return;
    int s = src[w], d = dst[w];
    float norm = dinv[s] * dinv[d];
    float v = h[s * 32 + lane] * norm;
    unsafeAtomicAdd(&out[d * 32 + lane], v);
}

__global__ void k_edge_scatter64(const int* __restrict__ src,
                                 const int* __restrict__ dst,
                                 const float* __restrict__ dinv,
                                 const float* __restrict__ h,
                                 float* out, int E) {
    int w = (blockIdx.x * blockDim.x + threadIdx.x) >> 5;
    int lane = threadIdx.x & 31;
    if (w >= E) return;
    int s = src[w], d = dst[w];
    float norm = dinv[s] * dinv[d];
    const float2* hp = (const float2*)(h + (size_t)s * 64);
    float2 v = hp[lane];
    unsafeAtomicAdd(&out[d * 64 + 2 * lane + 0], v.x * norm);
    unsafeAtomicAdd(&out[d * 64 + 2 * lane + 1], v.y * norm);
}

// ---------------- bias + ReLU (in place) ----------------

template <int C>
__global__ void k_bias_relu(float* h, const float* __restrict__ bias,
                            int total) {
    int id = blockIdx.x * blockDim.x + threadIdx.x;
    if (id >= total) return;
    float v = h[id] + bias[id & (C - 1)];
    h[id] = v > 0.0f ? v : 0.0f;
}

// ---------------- pooling ----------------

__global__ void k_pool_accum(const float* __restrict__ h,
                             const int* __restrict__ batch,
                             float* sums, int n) {
    int id = blockIdx.x * blockDim.x + threadIdx.x;
    if (id >= n * 64) return;
    int node = id >> 6, c = id & 63;
    int g = batch[node];
    unsafeAtomicAdd(&sums[g * 64 + c], h[id]);
}

__global__ void k_count_accum(const int* __restrict__ batch, float* cnts, int n) {
    int i = blockIdx.x * blockDim.x + threadIdx.x;
    if (i < n) unsafeAtomicAdd(&cnts[batch[i]], 1.0f);
}

// ---------------- final head: pooled + logits ----------------
// d_out layout: [G*2 logits][G*64 pooled]

__global__ void k_final_head(const float* __restrict__ sums,
                             const float* __restrict__ cnts,
                             const float* __restrict__ Wc,
                             const float* __restrict__ bc,
                             float* out, int G) {
    int g = threadIdx.x;
    if (g >= G) return;
    float c = cnts[g];
    c = c > 1.0f ? c : 1.0f;
    float inv = 1.0f / c;
    float l0 = bc[0], l1 = bc[1];
    for (int ch = 0; ch < 64; ++ch) {
        float p = sums[g * 64 + ch] * inv;
        out[G * 2 + g * 64 + ch] = p;          // pooled
        l0 = fmaf(p, Wc[ch * 2 + 0], l0);
        l1 = fmaf(p, Wc[ch * 2 + 1], l1);
    }
    out[g * 2 + 0] = l0;
    out[g * 2 + 1] = l1;
}

// ---------------------------------------------------------------------------

extern "C" void kernel_launch(void* const* d_in, const int* in_sizes, int n_in,
                              void* d_out, int out_size, void* d_ws, size_t ws_size,
                              hipStream_t stream) {
    const float* x   = (const float*)d_in[0];
    const float* W1  = (const float*)d_in[1];
    const float* b1  = (const float*)d_in[2];
    const float* W2  = (const float*)d_in[3];
    const float* b2  = (const float*)d_in[4];
    const float* Wc  = (const float*)d_in[5];
    const float* bc  = (const float*)d_in[6];
    const int*   ei  = (const int*)d_in[7];
    const int*   bat = (const int*)d_in[8];

    const int n = in_sizes[0] / IN_CH;     // 100000
    const int E = in_sizes[7] / 2;         // 1600000
    const int G = 64;

    const int* src = ei;
    const int* dst = ei + E;

    // workspace carve-out (all f32)
    float* ws   = (float*)d_ws;
    float* deg  = ws;                ws += n;
    float* dinv = ws;                ws += n;
    float* h1   = ws;                ws += (size_t)n * 32;
    float* o1   = ws;                ws += (size_t)n * 32;
    float* h2   = ws;                ws += (size_t)n * 64;
    float* o2   = ws;                ws += (size_t)n * 64;
    float* sums = ws;                ws += (size_t)G * 64;
    float* cnts = ws;                ws += G;

    const int B = 256;
    auto cdiv = [](long long a, long long b) { return (int)((a + b - 1) / b); };

    // normalization
    k_init_deg<<<cdiv(n, B), B, 0, stream>>>(deg, n);
    k_deg_accum<<<cdiv(E, B), B, 0, stream>>>(dst, deg, E);
    k_dinv<<<cdiv(n, B), B, 0, stream>>>(deg, dinv, n);

    // layer 1: h1 = x @ W1  (K=128, N=32), 4 waves per block
    const int tiles = (n + 15) / 16;
    k_gemm_wmma<128, 32><<<cdiv(tiles, 4), 128, 0, stream>>>(x, W1, h1, n, tiles);
    k_selfloop<5><<<cdiv((long long)n * 32, B), B, 0, stream>>>(h1, dinv, o1, n * 32);
    k_edge_scatter32<<<cdiv((long long)E * 32, B), B, 0, stream>>>(src, dst, dinv, h1, o1, E);
    k_bias_relu<32><<<cdiv((long long)n * 32, B), B, 0, stream>>>(o1, b1, n * 32);

    // layer 2: h2 = o1 @ W2  (K=32, N=64)
    k_gemm_wmma<32, 64><<<cdiv(tiles, 4), 128, 0, stream>>>(o1, W2, h2, n, tiles);
    k_selfloop<6><<<cdiv((long long)n * 64, B), B, 0, stream>>>(h2, dinv, o2, n * 64);
    k_edge_scatter64<<<cdiv((long long)E * 32, B), B, 0, stream>>>(src, dst, dinv, h2, o2, E);
    k_bias_relu<64><<<cdiv((long long)n * 64, B), B, 0, stream>>>(o2, b2, n * 64);

    // pooling + head
    hipMemsetAsync(sums, 0, (size_t)(G * 64 + G) * sizeof(float), stream);
    k_pool_accum<<<cdiv((long long)n * 64, B), B, 0, stream>>>(o2, bat, sums, n);
    k_count_accum<<<cdiv(n, B), B, 0, stream>>>(bat, cnts, n);
    k_final_head<<<1, 64, 0, stream>>>(sums, cnts, Wc, bc, (float*)d_out, G);
}